// POI_Embedding_64450279244454
// MI455X (gfx1250) — compile-verified
//
#include <hip/hip_runtime.h>
#include <cstdint>
#include <cstddef>

// ---------------------------------------------------------------------------
// Embedding gather + LayerNorm (B*S = 131072 rows, DIM = 128, f32).
// Memory-bound: ~129 MB moved -> ~5.6 us floor @ 23.3 TB/s. The 51.2 MB
// embedding table is L2-resident (192 MB L2). No GEMM -> no WMMA; the CDNA5
// path used here is the async DMA engine (global_load_async_to_lds_b128 /
// ASYNCcnt) for a double-buffered per-wave gather pipeline, wave32 layout
// (1 wave == 1 row == 32 lanes x float4), and non-temporal output stores so
// the 64 MB output stream does not evict the L2-resident table.
// ---------------------------------------------------------------------------

typedef float v4f __attribute__((ext_vector_type(4)));
typedef int   v4i __attribute__((ext_vector_type(4)));
typedef __attribute__((address_space(1))) v4i g_v4i;  // global
typedef __attribute__((address_space(3))) v4i l_v4i;  // LDS

constexpr int   kDim          = 128;
constexpr int   kWave         = 32;          // CDNA5 is wave32-only
constexpr int   kWavesPerBlk  = 8;           // 256 threads/block
constexpr int   kBlock        = kWave * kWavesPerBlk;
constexpr int   kRowsPerWave  = 4;           // amortize pipeline prologue
constexpr float kEps          = 1e-5f;

#if defined(__HIP_DEVICE_COMPILE__) &&                                        \
    __has_builtin(__builtin_amdgcn_global_load_async_to_lds_b128) &&          \
    __has_builtin(__builtin_amdgcn_s_wait_asynccnt)
#define USE_ASYNC_LDS 1
#else
#define USE_ASYNC_LDS 0
#endif

__device__ __forceinline__ void ln_compute_store(v4f v, v4f g4, v4f b4,
                                                 float* __restrict__ outrow,
                                                 int lane) {
  // Per-lane partial sums over this lane's 4 columns.
  float s  = v.x + v.y + v.z + v.w;
  float ss = v.x * v.x + v.y * v.y + v.z * v.z + v.w * v.w;
  // Wave32 butterfly reduction (ds_bpermute crossbar, no LDS memory).
#pragma unroll
  for (int m = kWave / 2; m >= 1; m >>= 1) {
    s  += __shfl_xor(s, m, kWave);
    ss += __shfl_xor(ss, m, kWave);
  }
  const float mu   = s * (1.0f / kDim);
  const float var  = ss * (1.0f / kDim) - mu * mu;
  const float rstd = rsqrtf(var + kEps);

  v4f o;
  o.x = (v.x - mu) * rstd * g4.x + b4.x;
  o.y = (v.y - mu) * rstd * g4.y + b4.y;
  o.z = (v.z - mu) * rstd * g4.z + b4.z;
  o.w = (v.w - mu) * rstd * g4.w + b4.w;

  // NT store: 64 MB write-once stream; keep L2 for the embedding table.
  __builtin_nontemporal_store(o, reinterpret_cast<v4f*>(outrow) + lane);
}

__global__ void __launch_bounds__(kBlock)
embed_layernorm_kernel(const int* __restrict__ x,          // [nrows] token ids
                       const float* __restrict__ tok_embed,// [VOCAB, 128]
                       const float* __restrict__ gamma,    // [128]
                       const float* __restrict__ beta,     // [128]
                       float* __restrict__ out,            // [nrows, 128]
                       int nrows) {
  const int lane = threadIdx.x & (kWave - 1);
  const int wid  = threadIdx.x >> 5;
  const int gwid = blockIdx.x * kWavesPerBlk + wid;
  const int nw   = gridDim.x * kWavesPerBlk;

  // gamma/beta are loop-invariant per lane (lane owns columns 4*lane..4*lane+3).
  const v4f g4 = reinterpret_cast<const v4f*>(gamma)[lane];
  const v4f b4 = reinterpret_cast<const v4f*>(beta)[lane];

#if USE_ASYNC_LDS
  // Per-wave double buffer: 2 x 512 B per wave -> 8 KB per block.
  __shared__ float lds[kWavesPerBlk * 2 * kDim];
  float* const buf0 = &lds[(wid * 2 + 0) * kDim];
  float* const buf1 = &lds[(wid * 2 + 1) * kDim];

  // Prologue: kick off async gather of this wave's first row.
  if (gwid < nrows) {
    const int tok = __builtin_nontemporal_load(&x[gwid]);
    const float* src = tok_embed + (size_t)tok * kDim + lane * 4;
    __builtin_amdgcn_global_load_async_to_lds_b128(
        (g_v4i*)src, (l_v4i*)(buf0 + lane * 4), /*offset=*/0, /*cpol=*/0);
  }

  int p = 0;
  for (int row = gwid; row < nrows; row += nw, p ^= 1) {
    float* const cur = p ? buf1 : buf0;
    float* const nxt = p ? buf0 : buf1;

    const int next = row + nw;
    if (next < nrows) {
      // Issue gather for the next row, then wait until only it is in flight:
      // async loads retire in order, so asynccnt<=1 => current row is in LDS.
      const int tok = __builtin_nontemporal_load(&x[next]);
      const float* src = tok_embed + (size_t)tok * kDim + lane * 4;
      __builtin_amdgcn_global_load_async_to_lds_b128(
          (g_v4i*)src, (l_v4i*)(nxt + lane * 4), /*offset=*/0, /*cpol=*/0);
      __builtin_amdgcn_s_wait_asynccnt(1);
    } else {
      __builtin_amdgcn_s_wait_asynccnt(0);
    }

    const v4f v = *reinterpret_cast<const v4f*>(cur + lane * 4);
    ln_compute_store(v, g4, b4, out + (size_t)row * kDim, lane);
  }
#else
  // Fallback (toolchains without the async builtins): direct b128 gather to
  // VGPRs; latency hidden by 8 waves/block across many blocks.
  for (int row = gwid; row < nrows; row += nw) {
    const int tok = __builtin_nontemporal_load(&x[row]);
    const v4f v =
        reinterpret_cast<const v4f*>(tok_embed + (size_t)tok * kDim)[lane];
    ln_compute_store(v, g4, b4, out + (size_t)row * kDim, lane);
  }
#endif
}

extern "C" void kernel_launch(void* const* d_in, const int* in_sizes, int n_in,
                              void* d_out, int out_size, void* d_ws,
                              size_t ws_size, hipStream_t stream) {
  (void)n_in; (void)d_ws; (void)ws_size; (void)out_size;

  const int*   x         = (const int*)d_in[0];    // token ids [B*S]
  const float* tok_embed = (const float*)d_in[1];  // [VOCAB, 128]
  const float* gamma     = (const float*)d_in[2];  // [128]
  const float* beta      = (const float*)d_in[3];  // [128]
  float*       out       = (float*)d_out;          // [B*S, 128]

  const int nrows = in_sizes[0];  // B*S = 131072
  const int rows_per_block = kWavesPerBlk * kRowsPerWave;  // 32
  const int grid = (nrows + rows_per_block - 1) / rows_per_block;  // 4096

  embed_layernorm_kernel<<<grid, kBlock, 0, stream>>>(x, tok_embed, gamma,
                                                      beta, out, nrows);
}